// ModelNew_50388556317009
// MI455X (gfx1250) — compile-verified
//
#include <hip/hip_runtime.h>
#include <stdint.h>

// Depthwise 3x3 conv, fp32, N=16 C=64 H=W=512 -> 510x510, VALID, no bias.
// Memory-bound (AI ~2.2 flop/byte) => target HBM roofline (~92us @ 23.3TB/s).
// CDNA5 path: Tensor Data Mover (tensor_load_to_lds) stages each 66x66 input
// tile into LDS with hardware OOB->zero edge padding; VALU FMAs do the math.

#define CCH 64
#define HH  512
#define WW  512
#define OH  510
#define OW  510
#define TILE 64
#define IT   66   // input tile edge (64 + 2 halo)

typedef __attribute__((ext_vector_type(4))) unsigned int u32x4;
typedef __attribute__((ext_vector_type(8))) int          i32x8;
typedef __attribute__((ext_vector_type(4))) int          i32x4;

__global__ __launch_bounds__(256) void dwconv3x3_tdm(
    const float* __restrict__ x, const float* __restrict__ wgt,
    float* __restrict__ out) {
  __shared__ float tile[IT * IT];

  const int tIdx = blockIdx.x;        // 0..63 : 8x8 output tiles
  const int tx = tIdx & 7;
  const int ty = tIdx >> 3;
  const int c  = blockIdx.y;
  const int n  = blockIdx.z;
  const int x0 = tx * TILE;           // tile origin in input coords
  const int y0 = ty * TILE;

#if __has_builtin(__builtin_amdgcn_tensor_load_to_lds)
  // ---- TDM stage: each wave DMAs a horizontal slab of the 66x66 tile ----
  {
    const int wv   = __builtin_amdgcn_readfirstlane((int)(threadIdx.x >> 5)); // 0..7
    const int ys   = wv * 9;                       // LDS slab start row
    const int rows = (ys + 9 <= IT) ? 9 : (IT - ys); // 9 rows (last wave: 3)
    const int gy   = y0 + ys;                      // global start row (<512)

    const uint64_t ga = (uint64_t)(uintptr_t)(x +
        ((size_t)(n * CCH + c) * HH + (size_t)gy) * WW + (size_t)x0);
    // LDS byte offset: low 32 bits of the flat shared address.
    const uint32_t laddr =
        (uint32_t)(uintptr_t)(&tile[ys * IT]);

    // OOB -> zero: declare the "tensor" as the remainder of the image from
    // the slab origin; reads past the right/bottom edge return 0 into LDS.
    const uint32_t tdim0 = (uint32_t)(WW - x0);    // elements
    const uint32_t tdim1 = (uint32_t)(HH - gy);
    const uint32_t til0  = IT;                     // 66 elements per row
    const uint32_t til1  = (uint32_t)rows;
    const uint64_t str0  = WW;                     // row stride (elements)
    const uint64_t str1  = (uint64_t)WW * HH;      // unused (tile_dim2 = 0)

    // D# group 0: count=1 | lds_addr | global_addr[56:0] | type=2 @ [127:126]
    u32x4 g0 = {
      1u,
      laddr,
      (uint32_t)ga,
      (uint32_t)(((ga >> 32) & 0x01FFFFFFu) | 0x80000000u)
    };
    // D# group 1: data_size=4B; dims/strides packed per ISA bit layout.
    i32x8 g1 = {
      (int)(2u << 16),                                           // data_size
      (int)((tdim0 & 0xFFFFu) << 16),                            // tdim0 lo
      (int)((tdim0 >> 16) | ((tdim1 & 0xFFFFu) << 16)),          // tdim0 hi|tdim1 lo
      (int)((tdim1 >> 16) | (til0 << 16)),                       // tdim1 hi|tile0
      (int)(til1),                                               // tile1 | tile2=0
      (int)(uint32_t)str0,                                       // str0[31:0]
      (int)((uint32_t)(str0 >> 32) | (((uint32_t)str1 & 0xFFFFu) << 16)),
      (int)(uint32_t)(str1 >> 16)
    };
    i32x4 gz = { 0, 0, 0, 0 };                                   // groups 2/3

#if __clang_major__ >= 23
    i32x8 gz8 = { 0, 0, 0, 0, 0, 0, 0, 0 };
    __builtin_amdgcn_tensor_load_to_lds(g0, g1, gz, gz, gz8, 0);
#else
    __builtin_amdgcn_tensor_load_to_lds(g0, g1, gz, gz, 0);
#endif
    __builtin_amdgcn_s_wait_tensorcnt(0);
  }
  __syncthreads();
#else
  // Fallback: cooperative fill (no TDM builtin available).
  for (int i = (int)threadIdx.x; i < IT * IT; i += 256) {
    const int ly = i / IT, lx = i - ly * IT;
    const int gy = y0 + ly, gx = x0 + lx;
    float v = 0.0f;
    if (gy < HH && gx < WW)
      v = x[((size_t)(n * CCH + c) * HH + gy) * WW + gx];
    tile[i] = v;
  }
  __syncthreads();
#endif

  // ---- Compute: each thread = 16-high x 1-wide output strip ----
  const float* wp = wgt + c * 9;
  const float w00 = wp[0], w01 = wp[1], w02 = wp[2];
  const float w10 = wp[3], w11 = wp[4], w12 = wp[5];
  const float w20 = wp[6], w21 = wp[7], w22 = wp[8];

  const int col  = (int)(threadIdx.x & 63);   // tile-local output col
  const int rsel = (int)(threadIdx.x >> 6);   // 0..3 -> 16-row strip
  const int oc   = x0 + col;
  const int or0  = y0 + rsel * 16;
  const bool colOK = (oc < OW);

  const int lbase = (rsel * 16) * IT + col;
  float a0 = tile[lbase],          a1 = tile[lbase + 1],      a2 = tile[lbase + 2];
  float b0 = tile[lbase + IT],     b1 = tile[lbase + IT + 1], b2 = tile[lbase + IT + 2];

  float* op = out + ((size_t)(n * CCH + c) * OH + or0) * OW + oc;

#pragma unroll
  for (int r = 0; r < 16; ++r) {
    const int li = lbase + (r + 2) * IT;
    const float c0 = tile[li], c1 = tile[li + 1], c2 = tile[li + 2];

    float acc =       w00 * a0;
    acc = fmaf(w01, a1, acc);
    acc = fmaf(w02, a2, acc);
    acc = fmaf(w10, b0, acc);
    acc = fmaf(w11, b1, acc);
    acc = fmaf(w12, b2, acc);
    acc = fmaf(w20, c0, acc);
    acc = fmaf(w21, c1, acc);
    acc = fmaf(w22, c2, acc);

    if (colOK && (or0 + r) < OH) op[(size_t)r * OW] = acc;

    a0 = b0; a1 = b1; a2 = b2;
    b0 = c0; b1 = c1; b2 = c2;
  }
}

extern "C" void kernel_launch(void* const* d_in, const int* in_sizes, int n_in,
                              void* d_out, int out_size, void* d_ws, size_t ws_size,
                              hipStream_t stream) {
  (void)in_sizes; (void)n_in; (void)out_size; (void)d_ws; (void)ws_size;
  const float* x = (const float*)d_in[0];
  const float* w = (const float*)d_in[1];
  float* out = (float*)d_out;

  dim3 grid(64, CCH, 16);   // 8x8 tiles, C, N
  dim3 block(256);
  dwconv3x3_tdm<<<grid, block, 0, stream>>>(x, w, out);
}